// SorscherRNN_59682865545235
// MI455X (gfx1250) — compile-verified
//
#include <hip/hip_runtime.h>
#include <hip/hip_bf16.h>
#include <math.h>

// ---------------------------------------------------------------------------
// Sorscher RNN on MI455X (gfx1250), fp32 end-to-end using V_WMMA_F32_16X16X4_F32
//   NG=4096, NP=512, B=64, T=100, ALPHA=0.1, BETA=0.5, relu non-negativity
//
// W_h (64 MB fp32) stays resident in the 192 MB L2 across all 100 steps;
// HBM traffic is only gs (~105 MB) + logits (~13 MB), trivial at 23.3 TB/s.
// Register-block over N so the A fragment is loaded once per 2-4 wmma.
// All GEMM strides are compile-time template params so global loads fold
// into base+immediate-offset form and pipeline (clause + partial loadcnt).
// ---------------------------------------------------------------------------

typedef float v2f __attribute__((ext_vector_type(2)));
typedef float v8f __attribute__((ext_vector_type(8)));

#define NG    4096
#define NP    512
#define BATCH 64
#define TSTEP 100
#define ALPHA 0.1f
#define BETA  0.5f

// D = A(16x4) x B(4x16) + C  with fp32 fragments:
//  A: lane l -> row l&15, K pair at 2*(l>>4)      (v2f)
//  B: lane l -> col l&15, K pair at 2*(l>>4)      (v2f)
//  C/D: d[r] -> M = r + 8*(l>>4), N = l&15        (v8f)
__device__ __forceinline__ v8f wmma4(v2f a, v2f b, v8f acc) {
  return __builtin_amdgcn_wmma_f32_16x16x4_f32(
      false, a, false, b, (short)0, acc, false, false);
}

// ---------------------------------------------------------------------------
// Generic GEMM: C[M,N] = A[M,K] @ B[N,K]^T ; one wave per 16x64 strip
// (4 N-tiles register-blocked: A fragment loaded once per 4 wmma).
// Strides/K are template constants -> immediate-offset loads, pipelined.
// grid covers tiles_m * (tiles_n/4) waves, 128 threads/block (4 waves).
// ---------------------------------------------------------------------------
template <int LDA, int LDB, int LDC, int K>
__global__ void wmma_gemm_kernel(const float* __restrict__ A,
                                 const float* __restrict__ B,
                                 float* __restrict__ C,
                                 int ngroups_n /* = tiles_n/4 */) {
  const int wid = blockIdx.x * (blockDim.x >> 5) + (threadIdx.x >> 5);
  const int tm  = wid / ngroups_n;
  const int tn0 = (wid % ngroups_n) * 4;
  const int lane = threadIdx.x & 31;
  const int half = lane >> 4;
  const int mn = lane & 15;

  const float* ap  = A + (size_t)(tm * 16 + mn) * LDA + 2 * half;
  const float* bp0 = B + (size_t)(tn0 * 16 + mn) * LDB + 2 * half;
  const float* bp1 = bp0 + (size_t)16 * LDB;
  const float* bp2 = bp1 + (size_t)16 * LDB;
  const float* bp3 = bp2 + (size_t)16 * LDB;

  v8f acc0 = {}, acc1 = {}, acc2 = {}, acc3 = {};
#pragma unroll 4
  for (int k = 0; k < K; k += 4) {
    const v2f a  = *(const v2f*)(ap + k);
    const v2f b0 = *(const v2f*)(bp0 + k);
    const v2f b1 = *(const v2f*)(bp1 + k);
    const v2f b2 = *(const v2f*)(bp2 + k);
    const v2f b3 = *(const v2f*)(bp3 + k);
    acc0 = wmma4(a, b0, acc0);
    acc1 = wmma4(a, b1, acc1);
    acc2 = wmma4(a, b2, acc2);
    acc3 = wmma4(a, b3, acc3);
  }

  float* crow = C + (size_t)(tm * 16 + 8 * half) * LDC + tn0 * 16 + mn;
#pragma unroll
  for (int r = 0; r < 8; ++r) {
    crow[(size_t)r * LDC +  0] = acc0[r];
    crow[(size_t)r * LDC + 16] = acc1[r];
    crow[(size_t)r * LDC + 32] = acc2[r];
    crow[(size_t)r * LDC + 48] = acc3[r];
  }
}

// ---------------------------------------------------------------------------
// RNN step: z = hid @ W_h^T + v_t @ W_in^T - BETA*vv ; update vv, z_prev;
// gs[:,t,:] = relu(z).
// Block = 128 threads (4 waves) = one (16 x 32) output strip:
//   2 N-tiles register-blocked, split-K by 4 waves, LDS reduction.
// grid = 4 m-tiles x 128 n-groups = 512 blocks -> 2048 waves.
// hid row m at hid + m*hlda (t==0: h0, lda=NG; t>0: gs slice, lda=T*NG)
// ---------------------------------------------------------------------------
__global__ void rnn_step_kernel(const float* __restrict__ hid, int hlda,
                                const float* __restrict__ Wh,
                                const float* __restrict__ v,
                                const float* __restrict__ Win,
                                float* __restrict__ vv,
                                float* __restrict__ zprev,
                                float* __restrict__ gs, int t) {
  __shared__ float red[4][2][256];

  const int blk = blockIdx.x;         // 0..511
  const int tm  = blk >> 7;           // 0..3   (batch tile)
  const int tn0 = (blk & 127) * 2;    // first of 2 Ng tiles
  const int w = threadIdx.x >> 5;     // split-K chunk id
  const int lane = threadIdx.x & 31;
  const int half = lane >> 4;
  const int mn = lane & 15;
  const int k0 = w * (NG / 4);

  const float* ap  = hid + (size_t)(tm * 16 + mn) * hlda + 2 * half + k0;
  const float* bp0 = Wh + (size_t)(tn0 * 16 + mn) * NG + 2 * half + k0;
  const float* bp1 = bp0 + (size_t)16 * NG;

  v8f acc0 = {}, acc1 = {};
#pragma unroll 8
  for (int k = 0; k < NG / 4; k += 4) {
    const v2f a  = *(const v2f*)(ap + k);
    const v2f b0 = *(const v2f*)(bp0 + k);
    const v2f b1 = *(const v2f*)(bp1 + k);
    acc0 = wmma4(a, b0, acc0);
    acc1 = wmma4(a, b1, acc1);
  }

#pragma unroll
  for (int r = 0; r < 8; ++r) {
    red[w][0][lane * 8 + r] = acc0[r];
    red[w][1][lane * 8 + r] = acc1[r];
  }
  __syncthreads();

  if (w == 0) {
#pragma unroll
    for (int i = 0; i < 2; ++i) {
      const int n = (tn0 + i) * 16 + mn;
      const v2f win = *(const v2f*)(Win + 2 * n);
#pragma unroll
      for (int r = 0; r < 8; ++r) {
        const int e = lane * 8 + r;
        float z = red[0][i][e] + red[1][i][e] + red[2][i][e] + red[3][i][e];
        const int m = tm * 16 + 8 * half + r;  // batch index
        const v2f vt = *(const v2f*)(v + ((size_t)m * TSTEP + t) * 2);
        const size_t idx = (size_t)m * NG + n;
        const float vvold = vv[idx];
        const float zpold = zprev[idx];
        z += vt.x * win.x + vt.y * win.y - BETA * vvold;
        float vvn = vvold + ALPHA * (zpold - vvold);
        vvn = fmaxf(vvn, 0.0f);  // NON_NEGATIVITY
        vv[idx] = vvn;
        zprev[idx] = z;
        gs[((size_t)m * TSTEP + t) * NG + n] = fmaxf(z, 0.0f);
      }
    }
  }
}

// ---------------------------------------------------------------------------
// Row-wise log_softmax in place: one wave32 per row of 512.
// ---------------------------------------------------------------------------
__global__ void logsoftmax_kernel(float* __restrict__ out, int rows, int cols) {
  const int wave = blockIdx.x * (blockDim.x >> 5) + (threadIdx.x >> 5);
  const int lane = threadIdx.x & 31;
  if (wave >= rows) return;
  float* row = out + (size_t)wave * cols;

  float m = -INFINITY;
  for (int j = lane; j < cols; j += 32) m = fmaxf(m, row[j]);
#pragma unroll
  for (int off = 16; off; off >>= 1) m = fmaxf(m, __shfl_xor(m, off, 32));

  float s = 0.0f;
  for (int j = lane; j < cols; j += 32) s += expf(row[j] - m);
#pragma unroll
  for (int off = 16; off; off >>= 1) s += __shfl_xor(s, off, 32);

  const float lse = m + logf(s);
  for (int j = lane; j < cols; j += 32) row[j] = row[j] - lse;
}

__global__ void zero_kernel(float* __restrict__ p, int n) {
  int i = blockIdx.x * blockDim.x + threadIdx.x;
  if (i < n) p[i] = 0.0f;
}

// ---------------------------------------------------------------------------
// Host-side orchestration (graph-capture safe: only kernel launches on stream)
// ---------------------------------------------------------------------------
extern "C" void kernel_launch(void* const* d_in, const int* in_sizes, int n_in,
                              void* d_out, int out_size, void* d_ws, size_t ws_size,
                              hipStream_t stream) {
  (void)in_sizes; (void)n_in; (void)out_size; (void)ws_size;

  const float* v     = (const float*)d_in[0];  // [B,T,2]
  const float* p0    = (const float*)d_in[1];  // [B,NP]
  const float* W_enc = (const float*)d_in[2];  // [NG,NP]
  const float* W_in  = (const float*)d_in[3];  // [NG,2]
  const float* W_h   = (const float*)d_in[4];  // [NG,NG]
  const float* W_dec = (const float*)d_in[5];  // [NP,NG]
  float* out = (float*)d_out;                  // [B,T,NP] log-softmax

  // Workspace layout (floats)
  float* ws = (float*)d_ws;
  float* h0 = ws;                           // B*NG
  float* vv = h0 + (size_t)BATCH * NG;      // B*NG
  float* zp = vv + (size_t)BATCH * NG;      // B*NG
  float* gs = zp + (size_t)BATCH * NG;      // B*T*NG = 26,214,400

  // 0) zero the recurrent state (vv, zprev adjacent -> one pass)
  {
    const int n = 2 * BATCH * NG;
    zero_kernel<<<(n + 255) / 256, 256, 0, stream>>>(vv, n);
  }

  // 1) encoder: h0[B,NG] = p0[B,NP] @ W_enc[NG,NP]^T
  //    tiles: 4 x 256 -> 4 * 64 = 256 waves -> 64 blocks
  wmma_gemm_kernel<NP, NP, NG, NP><<<64, 128, 0, stream>>>(
      p0, W_enc, h0, /*ngroups_n=*/(NG / 16) / 4);

  // 2) recurrence: 100 dependent step GEMMs; hidden state ping-pongs via gs
  for (int t = 0; t < TSTEP; ++t) {
    const float* hid = (t == 0) ? h0 : (gs + (size_t)(t - 1) * NG);
    const int hlda = (t == 0) ? NG : (TSTEP * NG);
    rnn_step_kernel<<<512, 128, 0, stream>>>(hid, hlda, W_h, v, W_in,
                                             vv, zp, gs, t);
  }

  // 3) decoder: out[B*T,NP] = gs[B*T,NG] @ W_dec[NP,NG]^T
  //    tiles: 400 x 32 -> 400 * 8 = 3200 waves -> 800 blocks
  wmma_gemm_kernel<NG, NG, NP, NG><<<800, 128, 0, stream>>>(
      gs, W_dec, out, /*ngroups_n=*/(NP / 16) / 4);

  // 4) log_softmax over last axis, in place on d_out (6400 rows of 512)
  logsoftmax_kernel<<<(BATCH * TSTEP) / 4, 128, 0, stream>>>(
      out, BATCH * TSTEP, NP);
}